// DiffusionModel2d_37830071943407
// MI455X (gfx1250) — compile-verified
//
#include <hip/hip_runtime.h>
#include <math.h>

// Batched SPD solve of the 5-point diffusion stencil via block-Thomas:
//   A (4096x4096, block-tridiagonal with 64x64 tridiagonal diagonal blocks and
//   DIAGONAL off-blocks) is never assembled. Per block-row we form
//   S_j = T_j - D_j G_{j-1} D_j and invert the dense 64x64 S_j with a
//   16-wide-panel Gauss-Jordan whose rank-16 updates are fp32 WMMA
//   (V_WMMA_F32_16X16X4_F32) tile-GEMMs. ~0.5 GFLOP total vs 0.37 TFLOP for
//   the reference dense LU. All per-step state lives in LDS; the 64 stored
//   inverses per batch (8.4 MB total, needed for back-substitution) stay
//   resident in the 192 MB L2.

#define BATCH 8
#define MG    64          // cells per dimension; n = MG*MG = 4096
#define NAV   65          // alpha vertices per dimension
#define WP    130         // padded pitch of the augmented [S | I] matrix
#define TPB   256         // 8 wave32s per workgroup (one workgroup per batch)

typedef float v2f __attribute__((ext_vector_type(2)));
typedef float v8f __attribute__((ext_vector_type(8)));

// D(16x16) += (negA ? -A : A)(16x16) * B(16x16), K split into 4 chained
// V_WMMA_F32_16X16X4_F32. A row-major at pitch ap, B row-major at pitch bp.
// A-operand layout (ISA 7.12.2): lane m=lane&15, VGPR0/1 hold K = 2*(lane>>4)
// and +1 within each K-quad. B mirrors A's K placement; C/D: VGPR r holds
// rows r (lanes 0-15) and r+8 (lanes 16-31).
__device__ __forceinline__ v8f gemm16_acc(const float* __restrict__ Ab, int ap, bool negA,
                                          const float* __restrict__ Bb, int bp, v8f acc)
{
    const int lane = threadIdx.x & 31;
    const int m    = lane & 15;
    const int kh   = (lane >> 4) << 1;   // 0 or 2
#pragma unroll
    for (int q = 0; q < 4; ++q) {
        const int k0 = 4 * q + kh;
        v2f a, bv;
        float a0 = Ab[m * ap + k0];
        float a1 = Ab[m * ap + k0 + 1];
        a.x = negA ? -a0 : a0;
        a.y = negA ? -a1 : a1;
        bv.x = Bb[k0 * bp + m];
        bv.y = Bb[(k0 + 1) * bp + m];
        acc = __builtin_amdgcn_wmma_f32_16x16x4_f32(false, a, false, bv,
                                                    (short)0, acc, false, false);
    }
    return acc;
}

__global__ __launch_bounds__(TPB)
void diffusion_block_thomas(const float* __restrict__ alpha,
                            const float* __restrict__ frhs,
                            float* __restrict__ out,
                            float* __restrict__ gws)
{
    __shared__ float sW[64][WP];        // [ S_j | I ] -> [ I | S_j^{-1} ]
    __shared__ float sY[64][64];        // forward-swept RHS y_j
    __shared__ float sPinv[16][17];     // inverted 16x16 pivot block
    __shared__ float sLb[3][16][17];    // staged multiplier blocks
    __shared__ float sKLj[64];          // K_left  row j   (== D_j)
    __shared__ float sKLj1[64];         // K_left  row j+1 (== D_{j+1})
    __shared__ float sKB[65];           // K_bottom/K_top edges of row j
    __shared__ float sPart[TPB];
    __shared__ float sVec[64];
    __shared__ float sU[64];

    const int b    = blockIdx.x;
    const int tid  = threadIdx.x;
    const int lane = tid & 31;
    const int wv   = tid >> 5;

    const float* A = alpha + (size_t)b * NAV * NAV;
    const float* F = frhs  + (size_t)b * MG * MG;
    float*       G = gws   + (size_t)b * 64 * 4096;   // S_j^{-1} archive

    // ====================== forward sweep over block rows ======================
    for (int j = 0; j < 64; ++j) {
        // edge diffusivities for this block row (recomputed, not cached: cheap)
        if (tid < 65) {
            sKB[tid] = __expf(0.5f * (A[j * NAV + tid] + A[(j + 1) * NAV + tid]));
        } else if (tid >= 128 && tid < 192) {
            int k = tid - 128;
            sKLj[k]  = __expf(0.5f * (A[j * NAV + k] + A[j * NAV + k + 1]));
        } else if (tid >= 192) {
            int k = tid - 192;
            sKLj1[k] = __expf(0.5f * (A[(j + 1) * NAV + k] + A[(j + 1) * NAV + k + 1]));
        }
        __syncthreads();

        // S_j = T_j - D_j G_{j-1} D_j  (G_{j-1} still lives in sW's right half)
        // fused with the matvec v = G_{j-1} y_{j-1}
        {
            const int   r   = tid >> 2;
            const int   c0  = (tid & 3) << 4;
            const float klr = sKLj[r];
            float part = 0.f;
#pragma unroll
            for (int cc = 0; cc < 16; ++cc) {
                const int c = c0 + cc;
                float s;
                if (c == r) {
                    s = sKLj[r] + sKLj1[r] + sKB[r] + sKB[r + 1];
                    if (j == 0) s += sKLj[r];          // Dirichlet term at j==0
                } else if (c == r - 1) s = -sKB[r];
                else if (c == r + 1)   s = -sKB[r + 1];
                else                   s = 0.f;
                if (j > 0) {
                    const float g = sW[r][64 + c];     // G_{j-1}[r][c]
                    s    -= klr * g * sKLj[c];
                    part += g * sY[j - 1][c];
                }
                sW[r][c] = s;
            }
            sPart[tid] = part;
        }
        __syncthreads();

        // y_j = f_j + D_j v ; reset right half of sW to identity
        if (tid < 64) {
            float v  = sPart[4 * tid] + sPart[4 * tid + 1]
                     + sPart[4 * tid + 2] + sPart[4 * tid + 3];
            float yv = F[(j << 6) + tid];
            if (j > 0) yv += sKLj[tid] * v;
            sY[j][tid] = yv;
        }
        for (int e = tid; e < 4096; e += TPB) {
            const int r = e >> 6, c = e & 63;
            sW[r][64 + c] = (r == c) ? 1.f : 0.f;
        }
        __syncthreads();

        // ---- Gauss-Jordan on [S | I], 16-wide panels, WMMA rank-16 updates ----
        for (int p = 0; p < 4; ++p) {
            const int base = p << 4;

            if (wv == 0) {
                // wave 0: invert the 16x16 SPD pivot block entirely in registers
                // (lane = one column of the augmented [P | I]); no barriers.
                float col[16];
#pragma unroll
                for (int r = 0; r < 16; ++r)
                    col[r] = (lane < 16) ? sW[base + r][base + lane]
                                         : ((lane - 16) == r ? 1.f : 0.f);
#pragma unroll
                for (int kk = 0; kk < 16; ++kk) {
                    float fac[16];
#pragma unroll
                    for (int r = 0; r < 16; ++r) fac[r] = __shfl(col[r], kk, 32);
                    const float inv = 1.f / fac[kk];
                    col[kk] *= inv;
#pragma unroll
                    for (int r = 0; r < 16; ++r)
                        if (r != kk) col[r] -= fac[r] * col[kk];
                }
                if (lane >= 16) {
#pragma unroll
                    for (int r = 0; r < 16; ++r) sPinv[r][lane - 16] = col[r];
                }
            } else {
                // waves 1-7: concurrently stage the three multiplier blocks
                for (int e = tid - 32; e < 768; e += (TPB - 32)) {
                    const int rbi = e >> 8, rem = e & 255;
                    const int r = rem >> 4, c = rem & 15;
                    const int rr = rbi + (rbi >= p);
                    sLb[rbi][r][c] = sW[(rr << 4) + r][base + c];
                }
            }
            __syncthreads();

            // pivot rows: W[p-rows, :] = Pinv * W[p-rows, :]  (1 col-tile/wave)
            {
                const int n  = lane & 15;
                const int rh = (lane >> 4) << 3;
                v8f acc = {0.f, 0.f, 0.f, 0.f, 0.f, 0.f, 0.f, 0.f};
                acc = gemm16_acc(&sPinv[0][0], 17, false,
                                 &sW[base][wv << 4], WP, acc);
#pragma unroll
                for (int r = 0; r < 8; ++r)
                    sW[base + rh + r][(wv << 4) + n] = acc[r];
            }
            __syncthreads();

            // trailing update: W[r-rows, :] -= L * W[p-rows, :]
            // 24 independent 16x16x16 tile jobs, exactly 3 per wave. The trip
            // count is a compile-time constant so no EXEC-masked loop structure
            // is generated around the WMMAs (EXEC statically all-ones).
#pragma unroll
            for (int it = 0; it < 3; ++it) {
                const int jb  = wv + (it << 3);
                const int rbi = jb >> 3;
                const int ct  = jb & 7;
                const int rr  = (rbi + (rbi >= p)) << 4;
                const int n   = lane & 15;
                const int rh  = (lane >> 4) << 3;
                const int cg  = (ct << 4) + n;
                v8f acc;
#pragma unroll
                for (int r = 0; r < 8; ++r) acc[r] = sW[rr + rh + r][cg];
                acc = gemm16_acc(&sLb[rbi][0][0], 17, true,
                                 &sW[base][ct << 4], WP, acc);
#pragma unroll
                for (int r = 0; r < 8; ++r) sW[rr + rh + r][cg] = acc[r];
            }
            __syncthreads();
        }

        // persist G_j = S_j^{-1} (right half) for the backward sweep (L2-resident)
        {
            float* gj = G + ((size_t)j << 12);
            for (int e = tid; e < 4096; e += TPB)
                gj[e] = sW[e >> 6][64 + (e & 63)];
        }
        __threadfence_block();
        __syncthreads();
    }

    // ====================== backward substitution ======================
    for (int j = 63; j >= 0; --j) {
        if (tid < 64) {
            float rv = sY[j][tid];
            if (j < 63) {
                const float kl1 = __expf(0.5f * (A[(j + 1) * NAV + tid]
                                               + A[(j + 1) * NAV + tid + 1]));
                rv += kl1 * sU[tid];                    // + D_{j+1} u_{j+1}
            }
            sVec[tid] = rv;
        }
        const float* gj = G + ((size_t)j << 12);
        if (j > 0)  // pull next inverse toward the WGP while we do the matvec
            __builtin_prefetch(G + ((size_t)(j - 1) << 12) + (tid << 4), 0, 0);
        __syncthreads();
        {
            const int r = tid >> 2, c0 = (tid & 3) << 4;
            float part = 0.f;
#pragma unroll
            for (int cc = 0; cc < 16; ++cc)
                part += gj[(r << 6) + c0 + cc] * sVec[c0 + cc];
            sPart[tid] = part;
        }
        __syncthreads();
        if (tid < 64) {
            const float u = sPart[4 * tid] + sPart[4 * tid + 1]
                          + sPart[4 * tid + 2] + sPart[4 * tid + 3];
            sU[tid] = u;
            out[((size_t)b << 12) + (j << 6) + tid] = u;
        }
        __syncthreads();
    }
}

extern "C" void kernel_launch(void* const* d_in, const int* in_sizes, int n_in,
                              void* d_out, int out_size, void* d_ws, size_t ws_size,
                              hipStream_t stream) {
    const float* alpha = (const float*)d_in[0];   // (8, 65, 65) f32
    const float* frhs  = (const float*)d_in[1];   // (8, 64, 64) f32
    float*       out   = (float*)d_out;           // (8, 64, 64) f32
    float*       gws   = (float*)d_ws;            // needs 8*64*4096*4 = 8.39 MB

    (void)in_sizes; (void)n_in; (void)out_size; (void)ws_size;
    diffusion_block_thomas<<<dim3(BATCH), dim3(TPB), 0, stream>>>(alpha, frhs, out, gws);
}